// L2BoundedLTICell_88845693485294
// MI455X (gfx1250) — compile-verified
//
#include <hip/hip_runtime.h>
#include <hip/hip_bf16.h>

typedef __attribute__((ext_vector_type(2))) float v2f;
typedef __attribute__((ext_vector_type(8))) float v8f;

#define BB   32      // batch
#define TT   4096    // time steps
#define DD   128     // feature dim
#define LL   32      // chunk length
#define CHN  128     // number of chunks (TT/LL)
#define LDX  132     // padded LDS row stride (floats) for A-fragment (stride-128) access

__device__ __forceinline__ v8f wmma4(v2f a, v2f b, v8f c) {
    return __builtin_amdgcn_wmma_f32_16x16x4_f32(
        /*neg_a=*/false, a, /*neg_b=*/false, b,
        /*c_mod=*/(short)0, c, /*reuse_a=*/false, /*reuse_b=*/false);
}

// Async 16B global -> LDS copy (ASYNCcnt path). LDS offset = low 32 bits of flat addr.
__device__ __forceinline__ void async_copy16(const void* gptr, void* lptr) {
    unsigned loff = (unsigned)(uintptr_t)lptr;
    unsigned long long gaddr = (unsigned long long)(uintptr_t)gptr;
    asm volatile("global_load_async_to_lds_b128 %0, %1, off"
                 :: "v"(loff), "v"(gaddr) : "memory");
}
__device__ __forceinline__ void async_wait0() {
    asm volatile("s_wait_asynccnt 0x0" ::: "memory");
}

// Packed B-fragment index: element pair for (n-tile nt, k-quad kq, lane)
__device__ __forceinline__ int pidx(int nt, int kq, int lane) {
    return ((nt * 32 + kq) * 32 + lane) * 2;
}

// ---------------------------------------------------------------------------
// Setup 1: power iteration for sigma = ||K_raw||_2; writes 1/(sigma+2e-3)
// ---------------------------------------------------------------------------
__global__ void sigma_kernel(const float* __restrict__ K, float* __restrict__ scale_out) {
    __shared__ float v[256], w[256], red[256];
    const int tid = threadIdx.x;
    v[tid] = 1.0f;
    __syncthreads();
    for (int it = 0; it < 200; ++it) {
        float acc = 0.f;
        const float* row = K + tid * 256;
        for (int j = 0; j < 256; ++j) acc += row[j] * v[j];
        w[tid] = acc;
        __syncthreads();
        float z = 0.f;
        for (int i = 0; i < 256; ++i) z += K[i * 256 + tid] * w[i];
        red[tid] = z * z;
        __syncthreads();
        for (int s = 128; s > 0; s >>= 1) {
            if (tid < s) red[tid] += red[tid + s];
            __syncthreads();
        }
        const float inv = rsqrtf(fmaxf(red[0], 1e-30f));
        __syncthreads();
        v[tid] = z * inv;
        __syncthreads();
    }
    float acc = 0.f;
    const float* row = K + tid * 256;
    for (int j = 0; j < 256; ++j) acc += row[j] * v[j];
    red[tid] = acc * acc;
    __syncthreads();
    for (int s = 128; s > 0; s >>= 1) {
        if (tid < s) red[tid] += red[tid + s];
        __syncthreads();
    }
    if (tid == 0) {
        float sg = sqrtf(red[0]);
        sg = fmaxf(sg, 1e-5f);
        scale_out[0] = 1.0f / (sg + 0.002f);
    }
}

// ---------------------------------------------------------------------------
// Setup 2: Gauss-Jordan inverse of S with partial pivoting (global scratch aug).
// ---------------------------------------------------------------------------
__global__ void gauss_kernel(const float* __restrict__ S, float* __restrict__ aug,
                             float* __restrict__ Sinv) {
    __shared__ float mx[128];
    __shared__ int   mi[128];
    __shared__ float fac[128];
    const int tid = threadIdx.x;
    for (int idx = tid; idx < 128 * 256; idx += 256) {
        const int r = idx >> 8, cc = idx & 255;
        aug[idx] = (cc < 128) ? S[r * 128 + cc] : ((cc - 128 == r) ? 1.f : 0.f);
    }
    __syncthreads();
    for (int p = 0; p < 128; ++p) {
        if (tid < 128) {
            mx[tid] = (tid >= p) ? fabsf(aug[tid * 256 + p]) : -1.f;
            mi[tid] = tid;
        }
        __syncthreads();
        for (int s = 64; s > 0; s >>= 1) {
            if (tid < s && mx[tid + s] > mx[tid]) { mx[tid] = mx[tid + s]; mi[tid] = mi[tid + s]; }
            __syncthreads();
        }
        const int piv = mi[0];
        __syncthreads();
        if (piv != p) {
            const float a = aug[p * 256 + tid], b = aug[piv * 256 + tid];
            aug[p * 256 + tid] = b;
            aug[piv * 256 + tid] = a;
        }
        __syncthreads();
        const float pv = aug[p * 256 + p];
        __syncthreads();
        aug[p * 256 + tid] = aug[p * 256 + tid] / pv;
        __syncthreads();
        if (tid < 128) fac[tid] = aug[tid * 256 + p];
        __syncthreads();
        const float pr = aug[p * 256 + tid];
        for (int r = 0; r < 128; ++r)
            if (r != p) aug[r * 256 + tid] -= fac[r] * pr;
        __syncthreads();
    }
    for (int idx = tid; idx < 128 * 128; idx += 256) {
        const int r = idx >> 7, cc = idx & 127;
        Sinv[idx] = aug[r * 256 + 128 + cc];
    }
}

// ---------------------------------------------------------------------------
// Small 128^3 matmul: C = s * A(lda) @ B(ldb).  <<<64,256>>>
// ---------------------------------------------------------------------------
__global__ void mm128_kernel(const float* __restrict__ A, int lda,
                             const float* __restrict__ B, int ldb,
                             float* __restrict__ C, const float* sptr) {
    const int idx = blockIdx.x * blockDim.x + threadIdx.x;
    const int i = idx >> 7, j = idx & 127;
    const float s = sptr ? sptr[0] : 1.0f;
    float acc = 0.f;
    for (int k = 0; k < 128; ++k) acc = fmaf(A[i * lda + k], B[k * ldb + j], acc);
    C[i * 128 + j] = s * acc;
}

__global__ void transpose_scale_kernel(const float* __restrict__ A, int lda,
                                       float* __restrict__ O, const float* sptr) {
    const int idx = blockIdx.x * blockDim.x + threadIdx.x;
    const int i = idx >> 7, j = idx & 127;
    const float s = sptr ? sptr[0] : 1.0f;
    O[j * 128 + i] = s * A[i * lda + j];
}

__global__ void ident_kernel(float* __restrict__ P) {
    const int idx = blockIdx.x * blockDim.x + threadIdx.x;
    P[idx] = ((idx >> 7) == (idx & 127)) ? 1.f : 0.f;
}

// G[j] = Bt @ P[L-1-j]   grid (64, LL)
__global__ void gmat_kernel(const float* __restrict__ Bt, const float* __restrict__ P,
                            float* __restrict__ G) {
    const int j = blockIdx.y;
    const int idx = blockIdx.x * blockDim.x + threadIdx.x;
    const int i = idx >> 7, n = idx & 127;
    const float* Pm = P + (size_t)(LL - 1 - j) * 16384;
    float acc = 0.f;
    for (int k = 0; k < 128; ++k) acc = fmaf(Bt[i * 128 + k], Pm[k * 128 + n], acc);
    G[(size_t)j * 16384 + i * 128 + n] = acc;
}

// ---------------------------------------------------------------------------
// Pass A: per chunk c, Q_c = sum_j U[c*L+j] @ G_j.
// G_j packed into LDS (fragment order) per j; u_t async-staged into LDS.
// Dynamic LDS: GP[16384] | US[32*LDX]
// ---------------------------------------------------------------------------
__global__ __launch_bounds__(512) void passA_kernel(const float* __restrict__ u,
                                                    const float* __restrict__ G,
                                                    float* __restrict__ Q) {
    extern __shared__ float smA[];
    float* GP = smA;             // 16384 packed floats
    float* US = smA + 16384;     // 32*LDX
    const int c = blockIdx.x;
    const int tid = threadIdx.x, lane = tid & 31, wid = tid >> 5;
    const int r0 = (wid >> 3) << 4, nt8 = wid & 7, n0 = nt8 << 4;
    const int m = lane & 15, klo = (lane >> 4) << 1, nn = lane & 15, half = lane >> 4;
    v8f acc = {0.f, 0.f, 0.f, 0.f, 0.f, 0.f, 0.f, 0.f};
    for (int j = 0; j < LL; ++j) {
        const size_t t = (size_t)c * LL + j;
        // pack G_j into fragment order (8192 pairs, 16 per thread)
        const float* g = G + (size_t)j * 16384;
        for (int pos = tid; pos < 8192; pos += 512) {
            const int pl = pos & 31, pkq = (pos >> 5) & 31, pnt = pos >> 10;
            const int pk = 4 * pkq + ((pl >> 4) << 1);
            const int pn = pnt * 16 + (pl & 15);
            GP[pos * 2]     = g[pk * 128 + pn];
            GP[pos * 2 + 1] = g[(pk + 1) * 128 + pn];
        }
        // async-stage u_t rows into LDS (1024 x 16B chunks, 2 per thread)
        for (int s = 0; s < 2; ++s) {
            const int q = tid + s * 512;
            const int b = q >> 5, cc = q & 31;
            async_copy16(u + ((size_t)b * TT + t) * DD + cc * 4, &US[b * LDX + cc * 4]);
        }
        async_wait0();
        __syncthreads();
#pragma unroll 4
        for (int kq = 0; kq < 32; ++kq) {
            const v2f a = *(const v2f*)&US[(r0 + m) * LDX + 4 * kq + klo];
            const v2f b = *(const v2f*)&GP[pidx(nt8, kq, lane)];
            acc = wmma4(a, b, acc);
        }
        __syncthreads();  // before GP/US overwrite next j
    }
    float* q = Q + (size_t)c * 4096;
#pragma unroll
    for (int i = 0; i < 8; ++i) q[(r0 + i + 8 * half) * 128 + n0 + nn] = acc[i];
}

// ---------------------------------------------------------------------------
// Pass B: sequential chunk-boundary states.  X_{c+1} = X_c @ P_L + Q_c.
// ---------------------------------------------------------------------------
__global__ __launch_bounds__(512) void passB_kernel(const float* __restrict__ x0,
                                                    const float* __restrict__ PL,
                                                    const float* __restrict__ Q,
                                                    float* __restrict__ XB) {
    __shared__ float Xs[2][BB * LDX];
    const int tid = threadIdx.x, lane = tid & 31, wid = tid >> 5;
    const int r0 = (wid >> 3) << 4, n0 = (wid & 7) << 4;
    const int m = lane & 15, klo = (lane >> 4) << 1, nn = lane & 15, half = lane >> 4;
    for (int i = tid; i < BB * DD; i += 512) Xs[0][(i >> 7) * LDX + (i & 127)] = x0[i];
    __syncthreads();
    int cur = 0;
    for (int c = 0; c < CHN; ++c) {
        float* xb = XB + (size_t)c * 4096;
        for (int i = tid; i < BB * DD; i += 512) xb[i] = Xs[cur][(i >> 7) * LDX + (i & 127)];
        const float* q = Q + (size_t)c * 4096;
        v8f acc;
#pragma unroll
        for (int i = 0; i < 8; ++i) acc[i] = q[(r0 + i + 8 * half) * 128 + n0 + nn];
#pragma unroll 4
        for (int k0 = 0; k0 < 128; k0 += 4) {
            v2f a; a.x = Xs[cur][(r0 + m) * LDX + k0 + klo];
                   a.y = Xs[cur][(r0 + m) * LDX + k0 + klo + 1];
            v2f b; b.x = PL[(k0 + klo) * 128 + n0 + nn];
                   b.y = PL[(k0 + klo + 1) * 128 + n0 + nn];
            acc = wmma4(a, b, acc);
        }
        const int nxt = cur ^ 1;
#pragma unroll
        for (int i = 0; i < 8; ++i) Xs[nxt][(r0 + i + 8 * half) * LDX + n0 + nn] = acc[i];
        __syncthreads();
        cur = nxt;
    }
}

// ---------------------------------------------------------------------------
// Pass C: per chunk, replay L steps. At/Ct fragments live in VGPRs (one-time
// preload), Bt/Dt packed in LDS, state double-buffered in LDS, u_t async-staged.
// Dynamic LDS: WPB[16384] | WPD[16384] | XS0[32*LDX] | XS1[32*LDX] | US[32*LDX]
// ---------------------------------------------------------------------------
__global__ __launch_bounds__(512) void passC_kernel(const float* __restrict__ u,
                                                    const float* __restrict__ At,
                                                    const float* __restrict__ Bt,
                                                    const float* __restrict__ Ct,
                                                    const float* __restrict__ Dt,
                                                    const float* __restrict__ XB,
                                                    float* __restrict__ yOut,
                                                    float* __restrict__ xOut) {
    extern __shared__ float smC[];
    float* WPB   = smC;                          // 16384
    float* WPD   = smC + 16384;                  // 16384
    float* XSbase = smC + 32768;                 // 2 * BB*LDX (double buffer)
    float* US    = smC + 32768 + 2 * BB * LDX;   // BB*LDX
    const int c = blockIdx.x;
    const int tid = threadIdx.x, lane = tid & 31, wid = tid >> 5;
    const int r0 = (wid >> 3) << 4, nt8 = wid & 7, n0 = nt8 << 4;
    const int m = lane & 15, klo = (lane >> 4) << 1, nn = lane & 15, half = lane >> 4;

    // one-time: pack Bt/Dt into LDS in fragment order (8192 pairs each)
    for (int pos = tid; pos < 8192; pos += 512) {
        const int pl = pos & 31, pkq = (pos >> 5) & 31, pnt = pos >> 10;
        const int pk = 4 * pkq + ((pl >> 4) << 1);
        const int pn = pnt * 16 + (pl & 15);
        WPB[pos * 2]     = Bt[pk * 128 + pn];
        WPB[pos * 2 + 1] = Bt[(pk + 1) * 128 + pn];
        WPD[pos * 2]     = Dt[pk * 128 + pn];
        WPD[pos * 2 + 1] = Dt[(pk + 1) * 128 + pn];
    }
    // one-time: boundary state into LDS (buffer 0)
    const float* xb = XB + (size_t)c * 4096;
    for (int i = tid; i < BB * DD; i += 512) XSbase[(i >> 7) * LDX + (i & 127)] = xb[i];
    // one-time: At/Ct fragments for this wave's n-tile into VGPRs (128 regs)
    v2f wA[32], wC[32];
#pragma unroll
    for (int kq = 0; kq < 32; ++kq) {
        const int k = 4 * kq + klo;
        wA[kq].x = At[k * 128 + n0 + nn];       wA[kq].y = At[(k + 1) * 128 + n0 + nn];
        wC[kq].x = Ct[k * 128 + n0 + nn];       wC[kq].y = Ct[(k + 1) * 128 + n0 + nn];
    }

    int cur = 0;
    for (int j = 0; j < LL; ++j) {
        const size_t t = (size_t)c * LL + j;
        float* Xcur = XSbase + cur * (BB * LDX);
        float* Xnxt = XSbase + (cur ^ 1) * (BB * LDX);
        // async-stage u_t into LDS
        for (int s = 0; s < 2; ++s) {
            const int q = tid + s * 512;
            const int b = q >> 5, cc = q & 31;
            async_copy16(u + ((size_t)b * TT + t) * DD + cc * 4, &US[b * LDX + cc * 4]);
        }
        async_wait0();
        __syncthreads();  // covers packing/boundary load (j==0) + US staging
        // x_seq records pre-update state
        for (int i = tid; i < BB * DD; i += 512) {
            const int b = i >> 7, d = i & 127;
            xOut[((size_t)b * TT + t) * DD + d] = Xcur[b * LDX + d];
        }
        v8f accY = {0.f, 0.f, 0.f, 0.f, 0.f, 0.f, 0.f, 0.f};
        v8f accX = {0.f, 0.f, 0.f, 0.f, 0.f, 0.f, 0.f, 0.f};
#pragma unroll
        for (int kq = 0; kq < 32; ++kq) {
            const v2f ax = *(const v2f*)&Xcur[(r0 + m) * LDX + 4 * kq + klo];
            const v2f au = *(const v2f*)&US[(r0 + m) * LDX + 4 * kq + klo];
            const v2f bB = *(const v2f*)&WPB[pidx(nt8, kq, lane)];
            const v2f bD = *(const v2f*)&WPD[pidx(nt8, kq, lane)];
            accY = wmma4(ax, wC[kq], accY);
            accY = wmma4(au, bD, accY);
            accX = wmma4(ax, wA[kq], accX);
            accX = wmma4(au, bB, accX);
        }
#pragma unroll
        for (int i = 0; i < 8; ++i) {
            const int brow = r0 + i + 8 * half;
            yOut[((size_t)brow * TT + t) * DD + n0 + nn] = accY[i];
            Xnxt[brow * LDX + n0 + nn] = accX[i];
        }
        __syncthreads();
        cur ^= 1;
    }
}

// ---------------------------------------------------------------------------
extern "C" void kernel_launch(void* const* d_in, const int* in_sizes, int n_in,
                              void* d_out, int out_size, void* d_ws, size_t ws_size,
                              hipStream_t stream) {
    const float* u    = (const float*)d_in[0];   // (32, 4096, 128)
    const float* x0   = (const float*)d_in[1];   // (32, 128)
    const float* S    = (const float*)d_in[2];   // (128, 128)
    const float* Kraw = (const float*)d_in[3];   // (256, 256)

    float* W = (float*)d_ws;
    float* SCALE = W;
    float* AUG   = SCALE + 16;
    float* SINV  = AUG   + 128 * 256;
    float* T1    = SINV  + 16384;
    float* AM    = T1    + 16384;
    float* BM    = AM    + 16384;
    float* CM    = BM    + 16384;
    float* AT    = CM    + 16384;
    float* BT    = AT    + 16384;
    float* CT    = BT    + 16384;
    float* DT    = CT    + 16384;
    float* P     = DT    + 16384;                     // (LL+1) * 16384
    float* G     = P     + (size_t)(LL + 1) * 16384;  // LL * 16384
    float* Q     = G     + (size_t)LL * 16384;        // CHN * 4096
    float* XB    = Q     + (size_t)CHN * 4096;        // CHN * 4096

    float* yOut = (float*)d_out;
    float* xOut = yOut + (size_t)BB * TT * DD;

    sigma_kernel<<<1, 256, 0, stream>>>(Kraw, SCALE);
    gauss_kernel<<<1, 256, 0, stream>>>(S, AUG, SINV);

    const float* K11 = Kraw;
    const float* K12 = Kraw + 128;
    const float* K21 = Kraw + (size_t)128 * 256;
    const float* K22 = Kraw + (size_t)128 * 256 + 128;
    mm128_kernel<<<64, 256, 0, stream>>>(SINV, 128, K11, 256, T1, SCALE);
    mm128_kernel<<<64, 256, 0, stream>>>(T1, 128, S, 128, AM, nullptr);
    mm128_kernel<<<64, 256, 0, stream>>>(SINV, 128, K12, 256, BM, SCALE);
    mm128_kernel<<<64, 256, 0, stream>>>(K21, 256, S, 128, CM, SCALE);
    transpose_scale_kernel<<<64, 256, 0, stream>>>(AM, 128, AT, nullptr);
    transpose_scale_kernel<<<64, 256, 0, stream>>>(BM, 128, BT, nullptr);
    transpose_scale_kernel<<<64, 256, 0, stream>>>(CM, 128, CT, nullptr);
    transpose_scale_kernel<<<64, 256, 0, stream>>>(K22, 256, DT, SCALE);

    ident_kernel<<<64, 256, 0, stream>>>(P);
    for (int k = 1; k <= LL; ++k)
        mm128_kernel<<<64, 256, 0, stream>>>(P + (size_t)(k - 1) * 16384, 128, AT, 128,
                                             P + (size_t)k * 16384, nullptr);
    gmat_kernel<<<dim3(64, LL), 256, 0, stream>>>(BT, P, G);

    const size_t smA_bytes = (size_t)(16384 + BB * LDX) * sizeof(float);
    const size_t smC_bytes = (size_t)(32768 + 3 * BB * LDX) * sizeof(float);
    passA_kernel<<<CHN, 512, smA_bytes, stream>>>(u, G, Q);
    passB_kernel<<<1, 512, 0, stream>>>(x0, P + (size_t)LL * 16384, Q, XB);
    passC_kernel<<<CHN, 512, smC_bytes, stream>>>(u, AT, BT, CT, DT, XB, yOut, xOut);
}